// grey_local_90872918049143
// MI455X (gfx1250) — compile-verified
//
#include <hip/hip_runtime.h>
#include <hip/hip_bf16.h>

#define NN 8192
#define BB 64
#define TT 12
#define EE 262144
#define NSTEP 12

typedef __attribute__((ext_vector_type(2))) float v2f;
typedef __attribute__((ext_vector_type(4))) float v4f;
typedef __attribute__((ext_vector_type(8))) float v8f;

// ---------------------------------------------------------------------------
// Per-node setup: total -> M -> 1/M, and explicit Ar_inv (12x12 row-major)
// by forward substitution (Ar is unit lower triangular with constant value
// v[j] along sub-diagonal row j).
// ---------------------------------------------------------------------------
__global__ void k_setup_node(const float* __restrict__ inputs,
                             const float* __restrict__ aggregator,
                             float* __restrict__ invM,   // N
                             float* __restrict__ Arinv)  // N*144
{
    int n = blockIdx.x * blockDim.x + threadIdx.x;
    if (n >= NN) return;
    const float FACT[TT] = {1.f, 1.f, 2.f, 6.f, 24.f, 120.f, 720.f,
                            5040.f, 40320.f, 362880.f, 3628800.f, 39916800.f};
    float r = aggregator[n];
    float v[TT];
    v[0] = 0.f;
    float p = 1.f;
    for (int j = 1; j < TT; ++j) { v[j] = p / FACT[j]; p *= (r + (float)(j - 1)); }

    // c_k = 1 + sum_{j>k} v_j  ;  total = (1/B) sum_k c_k * sum_b x[b,n,k]
    float c[TT];
    float s = 0.f;
    for (int k = TT - 1; k >= 0; --k) { c[k] = 1.f + s; s += v[k]; }
    float tot = 0.f;
    for (int k = 0; k < TT; ++k) {
        float sb = 0.f;
        for (int b = 0; b < BB; ++b)
            sb += inputs[(size_t)b * NN * TT + (size_t)n * TT + k];
        tot += c[k] * sb;
    }
    tot *= (1.0f / (float)BB);
    float M = (tot < 5.0f ? 5.0f : 0.0f) + fabsf(tot);
    invM[n] = 1.0f / M;

    float* An = Arinv + (size_t)n * TT * TT;
    for (int i = 0; i < TT * TT; ++i) An[i] = 0.f;
    for (int k = 0; k < TT; ++k) {
        An[k * TT + k] = 1.f;
        float S = 1.f;
        for (int j = k + 1; j < TT; ++j) {
            float z = -v[j] * S;
            An[j * TT + k] = z;
            S += z;
        }
    }
}

// x0 = last history sample, stored node-major: xcur[n*64 + b]
__global__ void k_init_x(const float* __restrict__ inputs, float* __restrict__ xcur)
{
    int i = blockIdx.x * blockDim.x + threadIdx.x;
    if (i >= NN * BB) return;
    int n = i >> 6, b = i & 63;
    xcur[i] = inputs[(size_t)b * NN * TT + (size_t)n * TT + (TT - 1)];
}

__global__ void k_zero(int* __restrict__ cnt)
{
    int i = blockIdx.x * blockDim.x + threadIdx.x;
    if (i < NN) cnt[i] = 0;
}

__global__ void k_hist(const int* __restrict__ dst, int* __restrict__ cnt)
{
    int e = blockIdx.x * blockDim.x + threadIdx.x;
    if (e < EE) atomicAdd(&cnt[dst[e]], 1);
}

// Single-block exclusive scan of 8192 counts -> row_ptr; resets cnt to 0.
__global__ void k_scan(int* __restrict__ cnt, int* __restrict__ row_ptr)
{
    __shared__ int sd[1024];
    int t = threadIdx.x;
    int base = t * 8;
    int loc[8];
    int s = 0;
    for (int i = 0; i < 8; ++i) { loc[i] = s; s += cnt[base + i]; }
    for (int i = 0; i < 8; ++i) cnt[base + i] = 0;
    sd[t] = s;
    __syncthreads();
    for (int off = 1; off < 1024; off <<= 1) {
        int v = (t >= off) ? sd[t - off] : 0;
        __syncthreads();
        sd[t] += v;
        __syncthreads();
    }
    int excl = sd[t] - s;
    for (int i = 0; i < 8; ++i) row_ptr[base + i] = excl + loc[i];
    if (t == 1023) row_ptr[NN] = sd[1023];
}

__global__ void k_scatter(const int* __restrict__ src, const int* __restrict__ dst,
                          const float* __restrict__ edge_w,
                          const int* __restrict__ row_ptr, int* __restrict__ cnt,
                          int* __restrict__ col, float* __restrict__ wcsr)
{
    int e = blockIdx.x * blockDim.x + threadIdx.x;
    if (e >= EE) return;
    int d = dst[e];
    int pos = atomicAdd(&cnt[d], 1);
    int o = row_ptr[d] + pos;
    col[o] = src[e];
    wcsr[o] = edge_w[e];
}

// ---------------------------------------------------------------------------
// One RK4 stage: k = f(xin); update accumulator / next stage input.
// One wave per node, float2 per lane (lane = batch pair): edge metadata goes
// through the scalar path (readfirstlane-uniform n), gathers are b64 loads,
// 256B coalesced per edge per wave.
// ---------------------------------------------------------------------------
__global__ void k_stage(const float* __restrict__ xin_, const float* __restrict__ xcur_,
                        float* __restrict__ acc_, float* __restrict__ xout_,
                        float* __restrict__ ysdst_,
                        const int* __restrict__ row_ptr, const int* __restrict__ col,
                        const float* __restrict__ wcsr,
                        const float* __restrict__ alpha, const float* __restrict__ invM,
                        int stage)
{
    int t = blockIdx.x * blockDim.x + threadIdx.x; // [0, N*32)
    int lane = t & 31;
    int n = __builtin_amdgcn_readfirstlane(t >> 5); // wave-uniform node id

    const v2f* xin  = (const v2f*)xin_;
    const v2f* xcur = (const v2f*)xcur_;
    v2f* acc  = (v2f*)acc_;
    v2f* xout = (v2f*)xout_;
    v2f* ys   = (v2f*)ysdst_;

    int idx = n * 32 + lane;
    v2f xs = xin[idx];

    int e0 = row_ptr[n], e1 = row_ptr[n + 1];
    float ex = 0.f, ey = 0.f;
    for (int e = e0; e < e1; ++e) {
        int sN  = col[e];   // scalar load (wave-uniform)
        float w = wcsr[e];  // scalar load (wave-uniform)
        v2f xv = xin[sN * 32 + lane];
        ex += w * (xv.x - xs.x);
        ey += w * (xv.y - xs.y);
    }
    float al = alpha[n], im = invM[n];
    float kx = ex + al * xs.x * (1.f - xs.x * im);
    float ky = ey + al * xs.y * (1.f - xs.y * im);

    const float dt = 1.0f / 12.0f;
    const float dt2 = dt * 0.5f, dt6 = dt / 6.0f, dt3 = dt / 3.0f;
    v2f o, a2;
    if (stage == 0) {                     // xin == xcur
        a2.x = xs.x + dt6 * kx;  a2.y = xs.y + dt6 * ky;
        o.x  = xs.x + dt2 * kx;  o.y  = xs.y + dt2 * ky;
        acc[idx] = a2;  xout[idx] = o;
    } else if (stage == 1) {
        v2f ac = acc[idx]; v2f xc = xcur[idx];
        ac.x += dt3 * kx;  ac.y += dt3 * ky;
        o.x = xc.x + dt2 * kx;  o.y = xc.y + dt2 * ky;
        acc[idx] = ac;  xout[idx] = o;
    } else if (stage == 2) {
        v2f ac = acc[idx]; v2f xc = xcur[idx];
        ac.x += dt3 * kx;  ac.y += dt3 * ky;
        o.x = xc.x + dt * kx;  o.y = xc.y + dt * ky;
        acc[idx] = ac;  xout[idx] = o;
    } else {
        v2f ac = acc[idx];
        o.x = ac.x + dt6 * kx;  o.y = ac.y + dt6 * ky;
        xout[idx] = o;          // xout == xcur (safe: no gathers from xcur here)
        ys[idx]   = o;
    }
}

// ---------------------------------------------------------------------------
// Output transform via WMMA: out[b,n,:] = Ar_inv[n] (12x12) @ Y[n] (12x64).
// One wave per node; A per ISA 16x4 f32 layout (lane=M, VGPR0/1 = K pair,
// lanes 16-31 at K+2); 3 K-chunks x 4 batch tiles = 12 v_wmma per node.
// D stored with b128: lanes 0-15 hold j=0..7 (two v4f), lanes 16-31 j=8..11.
// ---------------------------------------------------------------------------
__global__ void k_out_wmma(const float* __restrict__ Arinv,
                           const float* __restrict__ ysbuf,
                           float* __restrict__ out)
{
    int wave = threadIdx.x >> 5;
    int lane = threadIdx.x & 31;
    int node = blockIdx.x * 8 + wave;
    int m  = lane & 15;
    int hi = lane >> 4; // 0: K+0/K+1, 1: K+2/K+3

    const float* An = Arinv + (size_t)node * (TT * TT);
    float rowmask = (m < TT) ? 1.f : 0.f;
    int mr = (m < TT) ? m : 0;

    v2f a[3];
    for (int kc = 0; kc < 3; ++kc) {
        int K = kc * 4 + hi * 2; // K, K+1 in {0..11}
        a[kc].x = An[mr * TT + K] * rowmask;
        a[kc].y = An[mr * TT + K + 1] * rowmask;
    }

    for (int bt = 0; bt < 4; ++bt) {
        int batch = bt * 16 + m;
        v8f c = {};
        for (int kc = 0; kc < 3; ++kc) {
            int K = kc * 4 + hi * 2;
            v2f bf;
            bf.x = ysbuf[(size_t)K * NN * BB + (size_t)node * BB + batch];
            bf.y = ysbuf[(size_t)(K + 1) * NN * BB + (size_t)node * BB + batch];
            c = __builtin_amdgcn_wmma_f32_16x16x4_f32(
                    false, a[kc], false, bf, (short)0, c, false, false);
        }
        float* outp = out + (size_t)batch * NN * TT + (size_t)node * TT;
        v4f lo; lo.x = c[0]; lo.y = c[1]; lo.z = c[2]; lo.w = c[3];
        if (hi == 0) {
            v4f h4; h4.x = c[4]; h4.y = c[5]; h4.z = c[6]; h4.w = c[7];
            *(v4f*)(outp + 0) = lo;   // j = 0..3
            *(v4f*)(outp + 4) = h4;   // j = 4..7
        } else {
            *(v4f*)(outp + 8) = lo;   // j = 8..11
        }
    }
}

// ---------------------------------------------------------------------------
extern "C" void kernel_launch(void* const* d_in, const int* in_sizes, int n_in,
                              void* d_out, int out_size, void* d_ws, size_t ws_size,
                              hipStream_t stream)
{
    const float* inputs     = (const float*)d_in[0];
    const float* aggregator = (const float*)d_in[1];
    const float* edge_w     = (const float*)d_in[2];
    const float* alpha      = (const float*)d_in[3];
    const int*   src        = (const int*)d_in[4];
    const int*   dst        = (const int*)d_in[5];
    float* out = (float*)d_out;

    char* ws = (char*)d_ws;
    size_t cur = 0;
    auto take = [&](size_t bytes) {
        size_t off = cur;
        cur = (cur + bytes + 255) & ~(size_t)255;
        return off;
    };
    const size_t SB = (size_t)NN * BB * sizeof(float); // 2 MB state slab

    float* xcur  = (float*)(ws + take(SB));
    float* xsA   = (float*)(ws + take(SB));
    float* xsB   = (float*)(ws + take(SB));
    float* acc   = (float*)(ws + take(SB));
    float* ysbuf = (float*)(ws + take((size_t)NSTEP * SB));
    float* invM  = (float*)(ws + take((size_t)NN * sizeof(float)));
    float* Arinv = (float*)(ws + take((size_t)NN * TT * TT * sizeof(float)));
    int*   rowp  = (int*)  (ws + take((size_t)(NN + 1) * sizeof(int)));
    int*   cnt   = (int*)  (ws + take((size_t)NN * sizeof(int)));
    int*   col   = (int*)  (ws + take((size_t)EE * sizeof(int)));
    float* wcsr  = (float*)(ws + take((size_t)EE * sizeof(float)));
    (void)ws_size; (void)in_sizes; (void)n_in; (void)out_size;

    // Per-node constants + Ar_inv
    k_setup_node<<<NN / 256, 256, 0, stream>>>(inputs, aggregator, invM, Arinv);
    // x0
    k_init_x<<<(NN * BB) / 256, 256, 0, stream>>>(inputs, xcur);
    // CSR by destination
    k_zero<<<NN / 256, 256, 0, stream>>>(cnt);
    k_hist<<<EE / 256, 256, 0, stream>>>(dst, cnt);
    k_scan<<<1, 1024, 0, stream>>>(cnt, rowp);
    k_scatter<<<EE / 256, 256, 0, stream>>>(src, dst, edge_w, rowp, cnt, col, wcsr);

    // 12 RK4 steps, 4 stages each (ping-pong stage buffers; L2-resident state)
    const int SG = (NN * 32) / 256;
    for (int s = 0; s < NSTEP; ++s) {
        k_stage<<<SG, 256, 0, stream>>>(xcur, xcur, acc, xsA, nullptr,
                                        rowp, col, wcsr, alpha, invM, 0);
        k_stage<<<SG, 256, 0, stream>>>(xsA, xcur, acc, xsB, nullptr,
                                        rowp, col, wcsr, alpha, invM, 1);
        k_stage<<<SG, 256, 0, stream>>>(xsB, xcur, acc, xsA, nullptr,
                                        rowp, col, wcsr, alpha, invM, 2);
        k_stage<<<SG, 256, 0, stream>>>(xsA, xcur, acc, xcur,
                                        ysbuf + (size_t)s * NN * BB,
                                        rowp, col, wcsr, alpha, invM, 3);
    }

    // out = Ar_inv @ y  via v_wmma_f32_16x16x4_f32
    k_out_wmma<<<NN / 8, 256, 0, stream>>>(Arinv, ysbuf, out);
}